// LuongAttention_22110491640157
// MI455X (gfx1250) — compile-verified
//
#include <hip/hip_runtime.h>
#include <hip/hip_bf16.h>

#define BATCH  32
#define HIDDEN 1024
#define SRC    4096

typedef float v2f __attribute__((ext_vector_type(2)));
typedef float v8f __attribute__((ext_vector_type(8)));

// ---------------------------------------------------------------------------
// Kernel 1: proj = query @ W.T  (32x1024x1024 GEMM) via V_WMMA_F32_16X16X4_F32
// A = query (M=32 x K=1024), B[k][n] = W[n][k] (K=1024 x N=1024).
// One wave per 16x16 output tile; 4 waves/block over 4 N-tiles.
// A 16x4 layout : lane L holds A[M=L%16][k + 2*(L>=16) + {0,1}] in 2 VGPRs.
// B 4x16 layout : lane L holds B[k + 2*(L>=16) + {0,1}][N=L%16].
// C/D 16x16     : lane L, VGPR v -> M = v + 8*(L>=16), N = L%16.
// ---------------------------------------------------------------------------
__global__ __launch_bounds__(128) void proj_wmma_kernel(
    const float* __restrict__ q, const float* __restrict__ Wm,
    float* __restrict__ proj) {
  const int lane = threadIdx.x & 31;
  const int wave = threadIdx.x >> 5;
  const int l16  = lane & 15;
  const int half = lane >> 4;                       // 0 or 1
  const int mBase = blockIdx.x * 16;                // 2 tiles over M=32
  const int nBase = (blockIdx.y * 4 + wave) * 16;   // 64 tiles over N=1024

  const float* arow = q  + (size_t)(mBase + l16) * HIDDEN;
  const float* brow = Wm + (size_t)(nBase + l16) * HIDDEN;  // B[k][n] = W[n][k]

  v8f acc = {};
  for (int k = 0; k < HIDDEN; k += 4) {
    const int ka = k + 2 * half;
    v2f a, b;
    a.x = arow[ka];     a.y = arow[ka + 1];
    b.x = brow[ka];     b.y = brow[ka + 1];
    acc = __builtin_amdgcn_wmma_f32_16x16x4_f32(
        /*neg_a=*/false, a, /*neg_b=*/false, b,
        /*c_mod=*/(short)0, acc, /*reuse_a=*/false, /*reuse_b=*/false);
  }

#pragma unroll
  for (int v = 0; v < 8; ++v) {
    const int mm = mBase + v + 8 * half;
    proj[(size_t)mm * HIDDEN + nBase + l16] = acc[v];
  }
}

// ---------------------------------------------------------------------------
// Kernel 2: scores[b,s] = keys[b,s,:] . proj[b,:]  (+ mask -> -inf)
// One wave32 per s-row; proj row staged in LDS; float4 coalesced key loads;
// wave-level shuffle reduction.
// ---------------------------------------------------------------------------
__global__ __launch_bounds__(256) void scores_kernel(
    const float* __restrict__ keys, const float* __restrict__ proj,
    const unsigned char* __restrict__ mask, float* __restrict__ scores) {
  const int b    = blockIdx.x;
  const int wave = threadIdx.x >> 5;
  const int lane = threadIdx.x & 31;
  const int s    = blockIdx.y * 8 + wave;

  __shared__ float sproj[HIDDEN];
  for (int i = threadIdx.x; i < HIDDEN; i += 256)
    sproj[i] = proj[(size_t)b * HIDDEN + i];
  __syncthreads();

  const float* krow = keys + ((size_t)b * SRC + s) * HIDDEN;
  float acc = 0.0f;
#pragma unroll
  for (int j = 0; j < 8; ++j) {
    const int h = j * 128 + lane * 4;
    const float4 kv = *(const float4*)(krow + h);
    const float4 pv = *(const float4*)(sproj + h);
    acc = fmaf(kv.x, pv.x, acc);
    acc = fmaf(kv.y, pv.y, acc);
    acc = fmaf(kv.z, pv.z, acc);
    acc = fmaf(kv.w, pv.w, acc);
  }
#pragma unroll
  for (int off = 16; off > 0; off >>= 1)
    acc += __shfl_xor(acc, off, 32);

  if (lane == 0) {
    const size_t idx = (size_t)b * SRC + s;
    scores[idx] = mask[idx] ? -__builtin_inff() : acc;
  }
}

// ---------------------------------------------------------------------------
// Kernel 3: softmax over s per batch; writes attention weights (output #2).
// ---------------------------------------------------------------------------
__global__ __launch_bounds__(256) void softmax_kernel(
    const float* __restrict__ scores, float* __restrict__ attn) {
  const int b   = blockIdx.x;
  const int tid = threadIdx.x;
  __shared__ float red[256];

  const float* srow = scores + (size_t)b * SRC;
  float*       arow = attn   + (size_t)b * SRC;

  float m = -__builtin_inff();
  for (int i = tid; i < SRC; i += 256) m = fmaxf(m, srow[i]);
  red[tid] = m;
  __syncthreads();
  for (int o = 128; o > 0; o >>= 1) {
    if (tid < o) red[tid] = fmaxf(red[tid], red[tid + o]);
    __syncthreads();
  }
  const float gmax = red[0];
  __syncthreads();

  float sum = 0.0f;
  for (int i = tid; i < SRC; i += 256) {
    const float e = __expf(srow[i] - gmax);
    arow[i] = e;
    sum += e;
  }
  red[tid] = sum;
  __syncthreads();
  for (int o = 128; o > 0; o >>= 1) {
    if (tid < o) red[tid] += red[tid + o];
    __syncthreads();
  }
  const float inv = 1.0f / red[0];
  for (int i = tid; i < SRC; i += 256) arow[i] *= inv;
}

// ---------------------------------------------------------------------------
// Kernel 4: context partials. Grid (b, hChunk=4, sChunk=4); each block owns
// 256 h-columns and 1024 s-rows; attn chunk staged in LDS; values streamed
// fully coalesced (4 KB per iteration per block) with prefetch.
// ---------------------------------------------------------------------------
__global__ __launch_bounds__(256) void context_partial_kernel(
    const float* __restrict__ attn, const float* __restrict__ values,
    float* __restrict__ partial) {
  const int b  = blockIdx.x;
  const int hc = blockIdx.y;
  const int sc = blockIdx.z;
  const int h  = hc * 256 + threadIdx.x;
  const int s0 = sc * 1024;

  __shared__ float sa[1024];
  for (int i = threadIdx.x; i < 1024; i += 256)
    sa[i] = attn[(size_t)b * SRC + s0 + i];
  __syncthreads();

  const float* vp = values + ((size_t)b * SRC + s0) * HIDDEN + h;
  float acc = 0.0f;
  for (int si = 0; si < 1024; si += 4) {
    __builtin_prefetch(vp + (size_t)(si + 16) * HIDDEN, 0, 0);
    acc = fmaf(sa[si + 0], vp[(size_t)(si + 0) * HIDDEN], acc);
    acc = fmaf(sa[si + 1], vp[(size_t)(si + 1) * HIDDEN], acc);
    acc = fmaf(sa[si + 2], vp[(size_t)(si + 2) * HIDDEN], acc);
    acc = fmaf(sa[si + 3], vp[(size_t)(si + 3) * HIDDEN], acc);
  }
  partial[((size_t)sc * BATCH + b) * HIDDEN + h] = acc;
}

// ---------------------------------------------------------------------------
// Kernel 5: deterministic fixed-order reduce of the 4 s-chunk partials.
// ---------------------------------------------------------------------------
__global__ __launch_bounds__(256) void context_reduce_kernel(
    const float* __restrict__ partial, float* __restrict__ ctx) {
  const int i = blockIdx.x * 256 + threadIdx.x;  // 0 .. 32767
  float acc = partial[i];
  acc += partial[1 * BATCH * HIDDEN + i];
  acc += partial[2 * BATCH * HIDDEN + i];
  acc += partial[3 * BATCH * HIDDEN + i];
  ctx[i] = acc;
}

extern "C" void kernel_launch(void* const* d_in, const int* in_sizes, int n_in,
                              void* d_out, int out_size, void* d_ws, size_t ws_size,
                              hipStream_t stream) {
  const float*         query  = (const float*)d_in[0];
  const float*         keys   = (const float*)d_in[1];
  const float*         values = (const float*)d_in[2];
  const unsigned char* mask   = (const unsigned char*)d_in[3];
  const float*         Wm     = (const float*)d_in[4];

  float* ctx_out  = (float*)d_out;                  // (32, 1024)
  float* attn_out = (float*)d_out + BATCH * HIDDEN; // (32, 4096)

  float* proj    = (float*)d_ws;                 // 32*1024 floats
  float* scores  = proj + BATCH * HIDDEN;        // 32*4096 floats
  float* partial = scores;                       // reused after softmax (4*32*1024 floats)

  // 1) projection GEMM on the WMMA path
  proj_wmma_kernel<<<dim3(2, 16), 128, 0, stream>>>(query, Wm, proj);
  // 2) masked scores (streams 512 MB of keys)
  scores_kernel<<<dim3(BATCH, SRC / 8), 256, 0, stream>>>(keys, proj, mask, scores);
  // 3) softmax -> attention weights output
  softmax_kernel<<<dim3(BATCH), 256, 0, stream>>>(scores, attn_out);
  // 4) context partials (streams 512 MB of values)
  context_partial_kernel<<<dim3(BATCH, 4, 4), 256, 0, stream>>>(attn_out, values, partial);
  // 5) fixed-order reduce -> context output
  context_reduce_kernel<<<dim3((BATCH * HIDDEN) / 256), 256, 0, stream>>>(partial, ctx_out);
}